// OpnLayer_11974368821317
// MI455X (gfx1250) — compile-verified
//
#include <hip/hip_runtime.h>

typedef __attribute__((ext_vector_type(2))) float v2f;
typedef __attribute__((ext_vector_type(4))) float v4f;
typedef __attribute__((ext_vector_type(8))) float v8f;

#define BATCH 2048
#define FDIM  24
#define EDIM  128
#define PDIM  276          // F*(F-1)/2
#define LDS_STRIDE 132     // padded row stride (floats) -> 528B, bank-conflict free
#define BB_PER_WG 4        // batch blocks (of 128 rows) per workgroup; K_p stays LDS-resident

__global__ __launch_bounds__(256) void pair_bilinear_wmma(
    const float* __restrict__ x,      // [B, F, E]
    const float* __restrict__ kern,   // [E(d), P, E(e)]
    float* __restrict__ out)          // [B, P]
{
    __shared__ float sK[EDIM * LDS_STRIDE];   // K_p, padded: 67,584 B

    const int p       = blockIdx.y;
    const int b_super = blockIdx.x * (128 * BB_PER_WG);   // 512 rows per WG

    // map pair index p -> (fp, fq), triu(F, k=1) row-major order
    int fp = 0, rem = p, cnt = FDIM - 1;
    while (rem >= cnt) { rem -= cnt; ++fp; --cnt; }
    const int fq = fp + 1 + rem;

    const int tid = threadIdx.x;

    // --- stage K_p[d][e] = kern[d*P*E + p*E + e] into padded LDS (float4) ---
    const float* kp = kern + (size_t)p * EDIM;
    #pragma unroll
    for (int i = 0; i < 16; ++i) {
        int c  = tid + i * 256;               // 4096 float4 chunks
        int d  = c >> 5;                      // /32 chunks per row
        int e4 = (c & 31) << 2;
        v4f v = *(const v4f*)(kp + (size_t)d * (PDIM * EDIM) + e4);
        *(v4f*)(&sK[d * LDS_STRIDE + e4]) = v;
    }
    __syncthreads();

    const int wave = tid >> 5;
    const int lane = tid & 31;
    const int n16  = lane & 15;               // A row (M) for this lane / B col (N)
    const int kb   = (lane >> 4) * 2;         // K sub-offset: lanes 0-15 -> K0..1, 16-31 -> K2..3
    const int hi   = lane >> 4;

    // K_p is read-only in LDS from here on: loop batch blocks with it resident.
    #pragma unroll 1
    for (int bb = 0; bb < BB_PER_WG; ++bb) {
        const int brow = b_super + bb * 128 + wave * 16;

        // Xp row for A fragment: x[(brow + n16), fp, :]
        const float* xp_row = x + ((size_t)(brow + n16) * FDIM + fp) * EDIM;

        v8f acc[8];
        #pragma unroll
        for (int t = 0; t < 8; ++t)
            acc[t] = (v8f){0.f, 0.f, 0.f, 0.f, 0.f, 0.f, 0.f, 0.f};

        // --- T[m][d] = sum_e Xp[m][e] * K_p[d][e] ;  D = A(16xK=E) x B(KxN=d) ---
        #pragma unroll 4
        for (int e0 = 0; e0 < EDIM; e0 += 4) {
            v2f a = *(const v2f*)(xp_row + kb + e0);      // A frag: Xp[n16][kb+e0 .. +1]
            #pragma unroll
            for (int t = 0; t < 8; ++t) {
                // B frag: K_p[16t + n16][kb+e0 .. +1]  (ds_load b64, padded rows)
                v2f b = *(const v2f*)(&sK[(t * 16 + n16) * LDS_STRIDE + kb + e0]);
                acc[t] = __builtin_amdgcn_wmma_f32_16x16x4_f32(
                             false, a, false, b, (short)0, acc[t], false, false);
            }
        }

        // --- epilogue: result[m] = sum_d T[m][d] * Xq[m][d] ---
        // lane holds acc[t][r] = T[r + 8*hi][16*t + n16]
        #pragma unroll
        for (int r = 0; r < 8; ++r) {
            const int m = r + 8 * hi;
            const float* xq = x + ((size_t)(brow + m) * FDIM + fq) * EDIM;
            float s = 0.f;
            #pragma unroll
            for (int t = 0; t < 8; ++t)
                s += acc[t][r] * xq[t * 16 + n16];
            // reduce over the 16 lanes of this half (bit4 untouched -> halves separate)
            s += __shfl_xor(s, 1, 32);
            s += __shfl_xor(s, 2, 32);
            s += __shfl_xor(s, 4, 32);
            s += __shfl_xor(s, 8, 32);
            if (n16 == 0)
                out[(size_t)(brow + m) * PDIM + p] = s;
        }
    }
}

extern "C" void kernel_launch(void* const* d_in, const int* in_sizes, int n_in,
                              void* d_out, int out_size, void* d_ws, size_t ws_size,
                              hipStream_t stream) {
    const float* x    = (const float*)d_in[0];   // [2048, 24, 128]
    const float* kern = (const float*)d_in[1];   // [128, 276, 128]
    float* out = (float*)d_out;                  // [2048, 276]

    dim3 grid(BATCH / (128 * BB_PER_WG), PDIM);  // (4, 276)
    dim3 block(256);
    pair_bilinear_wmma<<<grid, block, 0, stream>>>(x, kern, out);
}